// SAGE_17721035063790
// MI455X (gfx1250) — compile-verified
//
#include <hip/hip_runtime.h>
#include <hip/hip_bf16.h>

// ---------------------------------------------------------------------------
// SAGEConv (pool aggregator) on MI455X / gfx1250, wave32.
//
// Problem sizes: B=2, N=10000, T=12, F_IN=F_OUT=64.  Rows R = B*N*T = 240000.
// All GEMMs are (R x 64) * (64 x 64) -> memory bound (5.9 GFLOP vs ~800MB
// traffic at 23.3 TB/s).  Therefore: stay in FP32 and use the exact-precision
// V_WMMA_F32_16X16X4_F32 matrix path (16 K-steps cover K=64).
//
// msg = relu(...) >= 0, so:
//   * neigh initialized to 0 == segment_max + zero-in-degree fixup
//   * scatter-max can use GLOBAL_ATOMIC_MAX_U32 on the float bit pattern
// ---------------------------------------------------------------------------

typedef __attribute__((ext_vector_type(2))) float v2f;
typedef __attribute__((ext_vector_type(8))) float v8f;

#define BB   2
#define NN   10000
#define TT   12
#define FF   64
#define EE   80000
#define ROWS (BB * NN * TT)     // 240000, divisible by 16
#define TF   (TT * FF)          // 768 floats per (b, node)

// ---------------------------------------------------------------------------
// Zero-fill (for neigh buffer)
// ---------------------------------------------------------------------------
__global__ void k_zero(float4* __restrict__ p, int n4) {
    int i = blockIdx.x * blockDim.x + threadIdx.x;
    if (i < n4) p[i] = make_float4(0.f, 0.f, 0.f, 0.f);
}

// ---------------------------------------------------------------------------
// Kernel 1: msg = relu(x @ W_pool^T + b_pool), rows in native (B,N,T,F) order.
// One wave handles a 16-row x 64-col output tile; 8 waves / block.
// ---------------------------------------------------------------------------
__global__ __launch_bounds__(256) void k_msg(const float* __restrict__ x,
                                             const float* __restrict__ Wp,
                                             const float* __restrict__ bp,
                                             float* __restrict__ msg) {
    const int lane   = threadIdx.x & 31;
    const int wave   = threadIdx.x >> 5;
    const int tile   = blockIdx.x * 8 + wave;
    const int R0     = tile * 16;                 // first row of this tile
    const int lane_m = lane & 15;                 // row/col within tile
    const int k_hi   = (lane >> 4) * 2;           // K sub-offset per half-wave

    const float* arow = x + (size_t)(R0 + lane_m) * FF;

    v8f c0 = {}, c1 = {}, c2 = {}, c3 = {};

    #pragma unroll
    for (int kk = 0; kk < 16; ++kk) {
        const int k0 = kk * 4 + k_hi;
        v2f a = *(const v2f*)(arow + k0);
        // B[k][g] = W_pool[g][k]; per-lane col g = n0 + lane_m, K = k0,k0+1
        v2f b0 = *(const v2f*)(Wp + (size_t)( 0 + lane_m) * FF + k0);
        v2f b1 = *(const v2f*)(Wp + (size_t)(16 + lane_m) * FF + k0);
        v2f b2 = *(const v2f*)(Wp + (size_t)(32 + lane_m) * FF + k0);
        v2f b3 = *(const v2f*)(Wp + (size_t)(48 + lane_m) * FF + k0);
        c0 = __builtin_amdgcn_wmma_f32_16x16x4_f32(false, a, false, b0, (short)0, c0, false, false);
        c1 = __builtin_amdgcn_wmma_f32_16x16x4_f32(false, a, false, b1, (short)0, c1, false, false);
        c2 = __builtin_amdgcn_wmma_f32_16x16x4_f32(false, a, false, b2, (short)0, c2, false, false);
        c3 = __builtin_amdgcn_wmma_f32_16x16x4_f32(false, a, false, b3, (short)0, c3, false, false);
    }

    // Epilogue: + b_pool, ReLU.  C/D layout: VGPR i -> row R0 + i + 8*(lane>=16),
    // col = n0 + lane_m.
    const int row_hi = (lane >> 4) * 8;
    #pragma unroll
    for (int nt = 0; nt < 4; ++nt) {
        const v8f c  = (nt == 0) ? c0 : (nt == 1) ? c1 : (nt == 2) ? c2 : c3;
        const int col = nt * 16 + lane_m;
        const float bv = bp[col];
        #pragma unroll
        for (int i = 0; i < 8; ++i) {
            float v = c[i] + bv;
            msg[(size_t)(R0 + row_hi + i) * FF + col] = v > 0.f ? v : 0.f;
        }
    }
}

// ---------------------------------------------------------------------------
// Kernel 2: per-edge scatter-max.  neigh[b,dst,:,:] = max(msg[b,src,:,:]).
// One block per edge (src/dst uniform -> SGPR); 384 threads cover B*T*F/4
// float4 elements.  msg >= 0 => unsigned-int atomic max == float max.
// ---------------------------------------------------------------------------
__global__ __launch_bounds__(384) void k_edgemax(const float* __restrict__ msg,
                                                 const int* __restrict__ src,
                                                 const int* __restrict__ dst,
                                                 unsigned int* __restrict__ neigh_u) {
    const int e = blockIdx.x;
    const int s = src[e];
    const int d = dst[e];
    const int i   = threadIdx.x;      // 0..383
    const int b   = i / 192;          // batch
    const int rem = i % 192;          // float4 index within (T*F)/4

    const float4 v = ((const float4*)(msg + ((size_t)(b * NN + s)) * TF))[rem];
    unsigned int* o = neigh_u + ((size_t)(b * NN + d)) * TF + rem * 4;
    atomicMax(o + 0, __float_as_uint(v.x));
    atomicMax(o + 1, __float_as_uint(v.y));
    atomicMax(o + 2, __float_as_uint(v.z));
    atomicMax(o + 3, __float_as_uint(v.w));
}

// ---------------------------------------------------------------------------
// Kernel 3: out = x @ W_self^T + neigh @ W_neigh^T + bias.
// Same WMMA tiling as k_msg; accumulators seeded with bias, two K-loops.
// ---------------------------------------------------------------------------
__global__ __launch_bounds__(256) void k_out(const float* __restrict__ x,
                                             const float* __restrict__ neigh,
                                             const float* __restrict__ Ws,
                                             const float* __restrict__ Wn,
                                             const float* __restrict__ bias,
                                             float* __restrict__ out) {
    const int lane   = threadIdx.x & 31;
    const int wave   = threadIdx.x >> 5;
    const int tile   = blockIdx.x * 8 + wave;
    const int R0     = tile * 16;
    const int lane_m = lane & 15;
    const int k_hi   = (lane >> 4) * 2;

    // Seed accumulators with bias (same column for every row of the tile).
    v8f c0, c1, c2, c3;
    {
        const float b0 = bias[ 0 + lane_m];
        const float b1 = bias[16 + lane_m];
        const float b2 = bias[32 + lane_m];
        const float b3 = bias[48 + lane_m];
        #pragma unroll
        for (int i = 0; i < 8; ++i) { c0[i] = b0; c1[i] = b1; c2[i] = b2; c3[i] = b3; }
    }

    const float* arow_h = x     + (size_t)(R0 + lane_m) * FF;
    const float* arow_n = neigh + (size_t)(R0 + lane_m) * FF;

    #pragma unroll
    for (int kk = 0; kk < 16; ++kk) {            // self term
        const int k0 = kk * 4 + k_hi;
        v2f a  = *(const v2f*)(arow_h + k0);
        v2f b0 = *(const v2f*)(Ws + (size_t)( 0 + lane_m) * FF + k0);
        v2f b1 = *(const v2f*)(Ws + (size_t)(16 + lane_m) * FF + k0);
        v2f b2 = *(const v2f*)(Ws + (size_t)(32 + lane_m) * FF + k0);
        v2f b3 = *(const v2f*)(Ws + (size_t)(48 + lane_m) * FF + k0);
        c0 = __builtin_amdgcn_wmma_f32_16x16x4_f32(false, a, false, b0, (short)0, c0, false, false);
        c1 = __builtin_amdgcn_wmma_f32_16x16x4_f32(false, a, false, b1, (short)0, c1, false, false);
        c2 = __builtin_amdgcn_wmma_f32_16x16x4_f32(false, a, false, b2, (short)0, c2, false, false);
        c3 = __builtin_amdgcn_wmma_f32_16x16x4_f32(false, a, false, b3, (short)0, c3, false, false);
    }

    #pragma unroll
    for (int kk = 0; kk < 16; ++kk) {            // neighbor term
        const int k0 = kk * 4 + k_hi;
        v2f a  = *(const v2f*)(arow_n + k0);
        v2f b0 = *(const v2f*)(Wn + (size_t)( 0 + lane_m) * FF + k0);
        v2f b1 = *(const v2f*)(Wn + (size_t)(16 + lane_m) * FF + k0);
        v2f b2 = *(const v2f*)(Wn + (size_t)(32 + lane_m) * FF + k0);
        v2f b3 = *(const v2f*)(Wn + (size_t)(48 + lane_m) * FF + k0);
        c0 = __builtin_amdgcn_wmma_f32_16x16x4_f32(false, a, false, b0, (short)0, c0, false, false);
        c1 = __builtin_amdgcn_wmma_f32_16x16x4_f32(false, a, false, b1, (short)0, c1, false, false);
        c2 = __builtin_amdgcn_wmma_f32_16x16x4_f32(false, a, false, b2, (short)0, c2, false, false);
        c3 = __builtin_amdgcn_wmma_f32_16x16x4_f32(false, a, false, b3, (short)0, c3, false, false);
    }

    const int row_hi = (lane >> 4) * 8;
    #pragma unroll
    for (int nt = 0; nt < 4; ++nt) {
        const v8f c  = (nt == 0) ? c0 : (nt == 1) ? c1 : (nt == 2) ? c2 : c3;
        const int col = nt * 16 + lane_m;
        #pragma unroll
        for (int i = 0; i < 8; ++i) {
            out[(size_t)(R0 + row_hi + i) * FF + col] = c[i];
        }
    }
}

// ---------------------------------------------------------------------------
// Launch.  Inputs (setup_inputs order): x, src, dst, W_pool, b_pool, W_neigh,
// W_self, bias.  Scratch: msg (61.44MB) + neigh (61.44MB) in d_ws.
// ---------------------------------------------------------------------------
extern "C" void kernel_launch(void* const* d_in, const int* in_sizes, int n_in,
                              void* d_out, int out_size, void* d_ws, size_t ws_size,
                              hipStream_t stream) {
    const float* x     = (const float*)d_in[0];
    const int*   src   = (const int*)  d_in[1];
    const int*   dst   = (const int*)  d_in[2];
    const float* Wp    = (const float*)d_in[3];
    const float* bp    = (const float*)d_in[4];
    const float* Wn    = (const float*)d_in[5];
    const float* Ws    = (const float*)d_in[6];
    const float* bias  = (const float*)d_in[7];
    float*       out   = (float*)d_out;

    const size_t nElem = (size_t)ROWS * FF;      // 15,360,000
    float* msg   = (float*)d_ws;
    float* neigh = msg + nElem;

    // 1) zero neigh (segment_max identity given relu(msg) >= 0)
    {
        const int n4 = (int)(nElem / 4);
        k_zero<<<(n4 + 255) / 256, 256, 0, stream>>>((float4*)neigh, n4);
    }

    const int tiles  = ROWS / 16;                // 15000
    const int blocks = tiles / 8;                // 1875 (8 waves/block)

    // 2) msg = relu(x @ W_pool^T + b_pool)
    k_msg<<<blocks, 256, 0, stream>>>(x, Wp, bp, msg);

    // 3) scatter-max over edges
    k_edgemax<<<EE, 384, 0, stream>>>(msg, src, dst, (unsigned int*)neigh);

    // 4) out = x @ W_self^T + neigh @ W_neigh^T + bias
    k_out<<<blocks, 256, 0, stream>>>(x, neigh, Ws, Wn, bias, out);
}